// LRIZZLoss_45775761441120
// MI455X (gfx1250) — compile-verified
//
#include <hip/hip_runtime.h>
#include <hip/hip_bf16.h>
#include <stdint.h>

// ---------------------------------------------------------------------------
// LRIZZ ranking loss for MI455X (gfx1250, wave32).
// Gather-latency-bound: 262k random 4B gathers from a 78.6MB fp32 tensor that
// is L2-resident (192MB L2). Per block: stage the contiguous 14336B targets
// chunk into LDS via the Tensor Data Mover (one tensor_load_to_lds, wave 0),
// overlap-friendly; gathers issued wide from 8 wave32 waves; deterministic
// two-kernel reduction; block partial written via async LDS->global store.
// ---------------------------------------------------------------------------

namespace {
constexpr int kB = 32, kC = 2, kH = 480, kW = 640, kN = 4096;
constexpr int kItems = kB * kN;                    // 131072 annotation pairs
constexpr int kTPB = 256;                          // 8 waves per block
constexpr int kItemsPerBlock = 512;
constexpr int kBlocks = kItems / kItemsPerBlock;   // 256
constexpr int kTgtInts = kItemsPerBlock * 7;       // 3584 dwords per block
constexpr int kIPT = kItemsPerBlock / kTPB;        // 2 items per thread
constexpr float kS = 1.0f, kL = 0.5f, kWEq = 1.0f, kWIneq = 1.0f;
static_assert(kN == 4096, "gid>>12 assumes N_ANNOS==4096");
static_assert(kTgtInts == 7 * 2 * kTPB, "unrolled async staging shape");
}  // namespace

#ifdef __HIP_DEVICE_COMPILE__
#define DEVPASS 1
#else
#define DEVPASS 0
#endif

#if DEVPASS && __has_builtin(__builtin_amdgcn_tensor_load_to_lds)
#define USE_TDM 1
#else
#define USE_TDM 0
#endif

#if DEVPASS && __has_builtin(__builtin_amdgcn_global_load_async_to_lds_b64)
#define USE_ASYNC_LOAD 1
#else
#define USE_ASYNC_LOAD 0
#endif

#if DEVPASS && __has_builtin(__builtin_amdgcn_global_store_async_from_lds_b128)
#define USE_ASYNC_STORE 1
#else
#define USE_ASYNC_STORE 0
#endif

#if DEVPASS && __has_builtin(__builtin_amdgcn_s_wait_asynccnt)
#define WAIT_ASYNC() __builtin_amdgcn_s_wait_asynccnt(0)
#elif DEVPASS
#define WAIT_ASYNC() asm volatile("s_wait_asynccnt 0x0" ::: "memory")
#else
#define WAIT_ASYNC() ((void)0)
#endif

#if DEVPASS && __has_builtin(__builtin_amdgcn_s_wait_tensorcnt)
#define WAIT_TENSOR() __builtin_amdgcn_s_wait_tensorcnt(0)
#elif DEVPASS
#define WAIT_TENSOR() asm volatile("s_wait_tensorcnt 0x0" ::: "memory")
#else
#define WAIT_TENSOR() ((void)0)
#endif

__global__ __launch_bounds__(kTPB) void lrizz_partial(
    const float* __restrict__ pred, const int* __restrict__ tgt,
    float* __restrict__ partial) {
  __shared__ __align__(16) int s_tgt[kTgtInts];
  __shared__ float s_red[kTPB / 32][4];
  __shared__ __align__(16) float s_out[4];

  const int tid = threadIdx.x;
  const int block_item0 = blockIdx.x * kItemsPerBlock;
  const int* gsrc = tgt + (size_t)block_item0 * 7;

  // ---- Stage this block's contiguous targets chunk (14336 B) into LDS ----
#if USE_TDM
  if (tid < 32) {  // wave-uniform: only wave 0 issues the DMA
    typedef unsigned int u32x4 __attribute__((ext_vector_type(4)));
    typedef int i32x4 __attribute__((ext_vector_type(4)));
    typedef int i32x8 __attribute__((ext_vector_type(8)));
    const uint64_t ga = (uint64_t)(uintptr_t)gsrc;
    const uint32_t lds = (uint32_t)(uintptr_t)&s_tgt[0];
    // D# group 0 (ISA 8.3): count=1 | lds_addr | global_addr[56:0] | type=2
    u32x4 g0;
    g0.x = 1u;                                   // count=1 (valid), user mode
    g0.y = lds;                                  // lds_addr (bytes)
    g0.z = (uint32_t)ga;                         // global_addr[31:0]
    g0.w = (uint32_t)((ga >> 32) & 0x1FFFFFFu) | (2u << 30);  // [56:32]|type=2
    // D# group 1 (ISA 8.4): 1D tile of kTgtInts dwords, data_size=4B (code 2),
    // no multicast / atomic-barrier / iterate / pad.
    i32x8 g1;
    g1[0] = 0x20000;                             // data_size=2 at bits[17:16]
    g1[1] = (int)(((unsigned)kTgtInts & 0xFFFFu) << 16);  // tensor_dim0[15:0]
    g1[2] = (int)(1u << 16);                     // tensor_dim0[31:16]=0, dim1=1
    g1[3] = (int)((unsigned)kTgtInts << 16);     // dim1[31:16]=0, tile_dim0
    g1[4] = 0;                                   // tile_dim1=0, tile_dim2=0
    g1[5] = kTgtInts;                            // tensor_dim0_stride[31:0]
    g1[6] = 0;                                   // stride0[47:32], stride1[15:0]
    g1[7] = 0;                                   // tensor_dim1_stride[47:16]
    i32x4 g2 = {0, 0, 0, 0};
    i32x4 g3 = {0, 0, 0, 0};
#if __clang_major__ >= 23
    i32x8 g4 = {0, 0, 0, 0, 0, 0, 0, 0};
    __builtin_amdgcn_tensor_load_to_lds(g0, g1, g2, g3, g4, 0);
#else
    __builtin_amdgcn_tensor_load_to_lds(g0, g1, g2, g3, 0);
#endif
    WAIT_TENSOR();
  }
#elif USE_ASYNC_LOAD
  {
    typedef int v2i __attribute__((ext_vector_type(2)));
    typedef __attribute__((address_space(1))) v2i GV2;
    typedef __attribute__((address_space(3))) v2i LV2;
    // 1792 b64 packets = exactly 7 per thread; constant strides fold into
    // the instruction's IOFFSET (applies to both global and LDS side).
    GV2* g = (GV2*)(uintptr_t)(gsrc + tid * 2);
    LV2* l = (LV2*)(uint32_t)(uintptr_t)(&s_tgt[tid * 2]);
#pragma unroll
    for (int k = 0; k < 7; ++k) {
      __builtin_amdgcn_global_load_async_to_lds_b64(g + k * kTPB, l + k * kTPB,
                                                    0, 0);
    }
    WAIT_ASYNC();
  }
#else
  for (int i = tid; i < kTgtInts; i += kTPB) s_tgt[i] = gsrc[i];
#endif
  __syncthreads();

  // ---- Compute gather indices for both items first (issue gathers wide) ---
  int idxA[kIPT], idxB[kIPT], lab[kIPT];
#pragma unroll
  for (int k = 0; k < kIPT; ++k) {
    const int li = tid + k * kTPB;
    const int gid = block_item0 + li;
    const int b = gid >> 12;  // gid / N_ANNOS
    const int* t = &s_tgt[li * 7];
    const int ca = t[0], xa = t[1], ya = t[2];
    const int cb = t[3], xb = t[4], yb = t[5];
    lab[k] = t[6];
    const int base = b * (kC * kH * kW);
    idxA[k] = base + (ca * kH + ya) * kW + xa;
    idxB[k] = base + (cb * kH + yb) * kW + xb;
  }

  float ineq_sum = 0.f, eq_sum = 0.f, ineq_cnt = 0.f, eq_cnt = 0.f;
#pragma unroll
  for (int k = 0; k < kIPT; ++k) {
    const float pa = pred[idxA[k]];
    const float pb = pred[idxB[k]];
    const float d = kS * (pb - pa);
    const float fl = (float)lab[k];
    // hinge: relu(S*L - S*diff*lab), S=1
    const float m = fmaxf(fmaf(-d, fl, kS * kL), 0.f);
    const bool is_ineq = (lab[k] != 0);
    ineq_sum += is_ineq ? m * m : 0.f;
    ineq_cnt += is_ineq ? 1.f : 0.f;
    eq_sum   += is_ineq ? 0.f : d * d;
    eq_cnt   += is_ineq ? 0.f : 1.f;
  }

  // ---- wave32 tree reduction (5 steps), then cross-wave via LDS ----------
#pragma unroll
  for (int off = 16; off > 0; off >>= 1) {
    ineq_sum += __shfl_down(ineq_sum, off, 32);
    eq_sum   += __shfl_down(eq_sum,   off, 32);
    ineq_cnt += __shfl_down(ineq_cnt, off, 32);
    eq_cnt   += __shfl_down(eq_cnt,   off, 32);
  }
  const int lane = tid & 31, wid = tid >> 5;
  if (lane == 0) {
    s_red[wid][0] = ineq_sum;
    s_red[wid][1] = eq_sum;
    s_red[wid][2] = ineq_cnt;
    s_red[wid][3] = eq_cnt;
  }
  __syncthreads();
  if (tid == 0) {
    float a = 0.f, b2 = 0.f, c = 0.f, d2 = 0.f;
#pragma unroll
    for (int w = 0; w < kTPB / 32; ++w) {
      a  += s_red[w][0];
      b2 += s_red[w][1];
      c  += s_red[w][2];
      d2 += s_red[w][3];
    }
#if USE_ASYNC_STORE
    // Result already lives in LDS: ship it straight LDS->global on the async
    // path (16B, single lane). S_ENDPGM's implicit wait-idle covers ASYNCcnt.
    s_out[0] = a; s_out[1] = b2; s_out[2] = c; s_out[3] = d2;
    typedef int v4i __attribute__((ext_vector_type(4)));
    typedef __attribute__((address_space(1))) v4i GV4;
    typedef __attribute__((address_space(3))) v4i LV4;
    GV4* g = (GV4*)(uintptr_t)&partial[(size_t)blockIdx.x * 4];
    LV4* l = (LV4*)(uint32_t)(uintptr_t)&s_out[0];
    __builtin_amdgcn_global_store_async_from_lds_b128(g, l, 0, 0);
#else
    (void)s_out;
    float4 r;
    r.x = a; r.y = b2; r.z = c; r.w = d2;
    *(float4*)&partial[(size_t)blockIdx.x * 4] = r;
#endif
  }
}

__global__ __launch_bounds__(kBlocks) void lrizz_finalize(
    const float* __restrict__ partial, float* __restrict__ out) {
  const int tid = threadIdx.x;
  const float4 p = *(const float4*)&partial[(size_t)tid * 4];
  float li = p.x, le = p.y, ci = p.z, ce = p.w;
#pragma unroll
  for (int off = 16; off > 0; off >>= 1) {
    li += __shfl_down(li, off, 32);
    le += __shfl_down(le, off, 32);
    ci += __shfl_down(ci, off, 32);
    ce += __shfl_down(ce, off, 32);
  }
  __shared__ float s[kBlocks / 32][4];
  const int lane = tid & 31, wid = tid >> 5;
  if (lane == 0) { s[wid][0] = li; s[wid][1] = le; s[wid][2] = ci; s[wid][3] = ce; }
  __syncthreads();
  if (tid == 0) {
    float tli = 0.f, tle = 0.f, tci = 0.f, tce = 0.f;
#pragma unroll
    for (int w = 0; w < kBlocks / 32; ++w) {
      tli += s[w][0]; tle += s[w][1]; tci += s[w][2]; tce += s[w][3];
    }
    // norm = num>0 ? 1/max(num,1) : 0  ==  num>0 ? 1/num : 0 (counts integral)
    const float ni = tci > 0.f ? 1.f / tci : 0.f;
    const float ne = tce > 0.f ? 1.f / tce : 0.f;
    out[0] = kWIneq * ni * tli + kWEq * ne * tle;
  }
}

extern "C" void kernel_launch(void* const* d_in, const int* in_sizes, int n_in,
                              void* d_out, int out_size, void* d_ws, size_t ws_size,
                              hipStream_t stream) {
  (void)in_sizes; (void)n_in; (void)out_size; (void)ws_size;
  const float* pred = (const float*)d_in[0];   // (32, 2, 480, 640) fp32
  const int* tgt = (const int*)d_in[1];        // (32, 4096, 7) int32
  float* partial = (float*)d_ws;               // kBlocks * 4 floats
  lrizz_partial<<<kBlocks, kTPB, 0, stream>>>(pred, tgt, partial);
  lrizz_finalize<<<1, kBlocks, 0, stream>>>(partial, (float*)d_out);
}